// GraphConvolutionalNetwork_84885733638246
// MI455X (gfx1250) — compile-verified
//
#include <hip/hip_runtime.h>

typedef __attribute__((ext_vector_type(2))) float v2f;
typedef __attribute__((ext_vector_type(8))) float v8f;

// ---------------------------------------------------------------- utilities

__global__ __launch_bounds__(256) void k_fill(float* __restrict__ p, float v, int n) {
    int i = blockIdx.x * 256 + threadIdx.x;
    if (i < n) p[i] = v;
}

__global__ __launch_bounds__(256) void k_degree(const long long* __restrict__ col,
                                                float* __restrict__ deg, int E) {
    int e = blockIdx.x * 256 + threadIdx.x;
    if (e < E) atomicAdd(deg + (int)col[e], 1.0f);
}

__global__ __launch_bounds__(256) void k_dinv(const float* __restrict__ deg,
                                              float* __restrict__ dinv, int n) {
    int i = blockIdx.x * 256 + threadIdx.x;
    if (i < n) dinv[i] = rsqrtf(deg[i]);   // deg >= 1 always (self-loop)
}

// ------------------------------------------------------- WMMA fp32 GEMM
// Y[M x N] = X[M x 64] @ W[64 x N], N in {64, 32}. One 16x16 tile per wave,
// K stepped by 4 via V_WMMA_F32_16X16X4_F32. 8 waves / 256-thread block.
// A frag (16x4): lane l%16 = M row; lane half h = lane/16 selects K pair
//   {kb+2h, kb+2h+1} -> contiguous v2f load (global_load_b64).
// B frag (4x16): lane l%16 = N col; VGPR j holds B[kb+2h+j][n].
// C/D (16x16): VGPR r holds D[m0 + r + 8h][n0 + l%16].
template<int N>
__global__ __launch_bounds__(256) void k_gemm_wmma(const float* __restrict__ X,
                                                   const float* __restrict__ W,
                                                   float* __restrict__ Y) {
    constexpr int K  = 64;
    constexpr int CT = N / 16;   // column tiles covered by the 8 waves
    constexpr int RT = 8 / CT;   // row tiles per block
    const int lane = threadIdx.x & 31;
    const int wave = threadIdx.x >> 5;
    const int ct   = wave % CT;
    const int rt   = wave / CT;
    const int m0   = blockIdx.x * (RT * 16) + rt * 16;
    const int l16  = lane & 15;
    const int h    = lane >> 4;
    const int n    = ct * 16 + l16;

    const float* xr = X + (size_t)(m0 + l16) * K;
    v8f acc = {};
#pragma unroll
    for (int kb = 0; kb < K; kb += 4) {
        const int ka = kb + 2 * h;
        v2f a = *(const v2f*)(xr + ka);          // A[m][ka], A[m][ka+1]
        v2f b;
        b.x = W[(size_t)(ka + 0) * N + n];       // B[ka][n]
        b.y = W[(size_t)(ka + 1) * N + n];       // B[ka+1][n]
        acc = __builtin_amdgcn_wmma_f32_16x16x4_f32(
            /*neg_a=*/false, a, /*neg_b=*/false, b,
            /*c_mod=*/(short)0, acc, /*reuse_a=*/false, /*reuse_b=*/false);
    }
#pragma unroll
    for (int r = 0; r < 8; ++r)
        Y[(size_t)(m0 + r + 8 * h) * N + n] = acc[r];
}

// ------------------------------------------------ edge gather + scatter-add
// agg[col[e]][:] += H[row[e]][:] * (dinv[row]*dinv[col]); one float4 chunk
// per thread, atomics resolve in L2 (accumulator fits in 192MB L2).
template<int F>
__global__ __launch_bounds__(256) void k_aggregate(const long long* __restrict__ rows,
                                                   const long long* __restrict__ cols,
                                                   const float* __restrict__ dinv,
                                                   const float* __restrict__ H,
                                                   float* __restrict__ AGG, int E) {
    constexpr int C4 = F / 4;
    int t = blockIdx.x * 256 + threadIdx.x;
    int e = t / C4;
    int c = t % C4;
    if (e >= E) return;
    int r  = (int)rows[e];
    int cl = (int)cols[e];
    float nrm = dinv[r] * dinv[cl];
    float4 hv = *(const float4*)(H + (size_t)r * F + 4 * c);
    float* dst = AGG + (size_t)cl * F + 4 * c;
    atomicAdd(dst + 0, hv.x * nrm);
    atomicAdd(dst + 1, hv.y * nrm);
    atomicAdd(dst + 2, hv.z * nrm);
    atomicAdd(dst + 3, hv.w * nrm);
}

// --------------------------------- self-loop + bias (+relu); OUT may == HL
template<int F, bool RELU>
__global__ __launch_bounds__(256) void k_finalize(const float* __restrict__ AGG,
                                                  const float* HL,
                                                  const float* __restrict__ dinv,
                                                  const float* __restrict__ bias,
                                                  float* OUT, int nNodes) {
    constexpr int C4 = F / 4;
    int t = blockIdx.x * 256 + threadIdx.x;
    int nd = t / C4;
    int c  = t % C4;
    if (nd >= nNodes) return;
    float d  = dinv[nd];
    float d2 = d * d;                       // self-loop norm
    float4 a  = *(const float4*)(AGG + (size_t)nd * F + 4 * c);
    float4 hv = *(const float4*)(HL  + (size_t)nd * F + 4 * c);
    float4 b  = *(const float4*)(bias + 4 * c);
    float4 o;
    o.x = a.x + hv.x * d2 + b.x;
    o.y = a.y + hv.y * d2 + b.y;
    o.z = a.z + hv.z * d2 + b.z;
    o.w = a.w + hv.w * d2 + b.w;
    if (RELU) {
        o.x = fmaxf(o.x, 0.f); o.y = fmaxf(o.y, 0.f);
        o.z = fmaxf(o.z, 0.f); o.w = fmaxf(o.w, 0.f);
    }
    *(float4*)(OUT + (size_t)nd * F + 4 * c) = o;
}

// --------------------------------------------------------------------------

extern "C" void kernel_launch(void* const* d_in, const int* in_sizes, int n_in,
                              void* d_out, int out_size, void* d_ws, size_t ws_size,
                              hipStream_t stream) {
    const float*     x   = (const float*)d_in[0];
    const long long* ei  = (const long long*)d_in[1];      // int64 [2, E]
    const float*     W1  = (const float*)d_in[3];
    const float*     b1  = (const float*)d_in[4];
    const float*     W2  = (const float*)d_in[5];
    const float*     b2  = (const float*)d_in[6];
    // edge_attr / We / be unused: _x_edge is not returned by the reference.

    const int Nn = in_sizes[0] / 64;     // 65536 nodes
    const int E  = in_sizes[1] / 2;      // 1048576 edges
    const long long* rows = ei;
    const long long* cols = ei + E;

    // Workspace layout (floats)
    float* deg  = (float*)d_ws;              // Nn
    float* dinv = deg  + Nn;                 // Nn
    float* h1   = dinv + Nn;                 // Nn*64   (becomes relu'd hidden)
    float* agg  = h1   + (size_t)Nn * 64;    // Nn*64   (reused for layer 2)
    float* h2   = agg  + (size_t)Nn * 64;    // Nn*32
    float* out  = (float*)d_out;             // Nn*32

    const int B = 256;
    // ---- degrees + normalization -------------------------------------
    k_fill<<<(Nn + B - 1) / B, B, 0, stream>>>(deg, 1.0f, Nn);     // self loops
    k_degree<<<(E + B - 1) / B, B, 0, stream>>>(cols, deg, E);
    k_dinv<<<(Nn + B - 1) / B, B, 0, stream>>>(deg, dinv, Nn);

    // ---- layer 1: h1 = x @ W1 ; agg = scatter ; relu(+b1) ------------
    k_gemm_wmma<64><<<Nn / 32, B, 0, stream>>>(x, W1, h1);
    k_fill<<<(Nn * 64 + B - 1) / B, B, 0, stream>>>(agg, 0.0f, Nn * 64);
    k_aggregate<64><<<(E * 16 + B - 1) / B, B, 0, stream>>>(rows, cols, dinv, h1, agg, E);
    k_finalize<64, true><<<(Nn * 16 + B - 1) / B, B, 0, stream>>>(agg, h1, dinv, b1, h1, Nn);

    // ---- layer 2: h2 = h1 @ W2 ; agg = scatter ; +b2 -----------------
    k_gemm_wmma<32><<<Nn / 64, B, 0, stream>>>(h1, W2, h2);
    k_fill<<<(Nn * 32 + B - 1) / B, B, 0, stream>>>(agg, 0.0f, Nn * 32);
    k_aggregate<32><<<(E * 8 + B - 1) / B, B, 0, stream>>>(rows, cols, dinv, h2, agg, E);
    k_finalize<32, false><<<(Nn * 8 + B - 1) / B, B, 0, stream>>>(agg, h2, dinv, b2, out, Nn);
}